// _UpFirDnNd_40149354283770
// MI455X (gfx1250) — compile-verified
//
#include <hip/hip_runtime.h>
#include <hip/hip_bf16.h>
#include <stdint.h>

// upfirdn2d: up=(2,2), down=(1,1), pad=((2,1),(2,1)), 4x4 kernel, gain=4
// in : (16,64,128,128) f32      out : (16,64,255,255) f32
// Each 2x2 output block (2by..2by+1, 2bx..2bx+1) depends on the 3x3 input
// patch centered at (by,bx) with zero border.

#define IN_H   128
#define IN_W   128
#define OUT_H  255
#define OUT_W  255
#define NTHR   256

typedef __attribute__((ext_vector_type(4))) unsigned int uint32x4_t;
typedef __attribute__((ext_vector_type(8))) int          int32x8_t;
typedef __attribute__((ext_vector_type(4))) int          int32x4_t;

struct F2u { float a, b; };   // 4-byte-aligned pair for odd-row stores

__device__ __forceinline__ float ldz(const float* s, int r, int c, bool ok) {
    // Load from a safe clamped address, select 0 when out of bounds.
    int rr = ok ? r : 0;
    int cc = ok ? c : 0;
    float v = s[(rr << 7) + cc];
    return ok ? v : 0.0f;
}

__global__ __launch_bounds__(NTHR)
void upfirdn2d_up2_k4(const float* __restrict__ x,
                      const float* __restrict__ kern,
                      float* __restrict__ out) {
    __shared__ float smem[IN_H * IN_W];   // 64 KiB: one (n,c) image

    const int img = blockIdx.x;                         // n*64 + c
    const float* src = x   + (size_t)img * (IN_H * IN_W);
    float*       dst = out + (size_t)img * (OUT_H * OUT_W);

    // Weights: W[4i+j] = K[i][j] times gain 4; flipped taps are picked at use.
    float W[16];
#pragma unroll
    for (int i = 0; i < 16; ++i) W[i] = kern[i] * 4.0f;

    // ---- Stage the image into LDS -------------------------------------
#if __has_builtin(__builtin_amdgcn_tensor_load_to_lds)
    if ((threadIdx.x >> 5) == 0) {   // wave 0 only (wave32)
        unsigned lds_off = (unsigned)(unsigned long long)(uintptr_t)&smem[0];
        unsigned long long ga = (unsigned long long)(uintptr_t)src;

        uint32x4_t g0;
        g0[0] = 1u;                                         // count=1, user D#
        g0[1] = lds_off;                                    // lds_addr
        g0[2] = (unsigned)(ga & 0xFFFFFFFFu);               // global_addr lo
        g0[3] = (unsigned)((ga >> 32) & 0x01FFFFFFu)        // global_addr hi
              | (2u << 30);                                 // type = 2 (image)

        int32x8_t g1;
        g1[0] = (2 << 16);          // workgroup_mask=0, data_size=2 (4B)
        g1[1] = (IN_W << 16);       // tensor_dim0[15:0]
        g1[2] = (IN_H << 16);       // tensor_dim0[31:16]=0, tensor_dim1[15:0]
        g1[3] = (IN_W << 16);       // tensor_dim1[31:16]=0, tile_dim0
        g1[4] = IN_H;               // tile_dim1, tile_dim2=0
        g1[5] = IN_W;               // tensor_dim0_stride[31:0]
        g1[6] = 0;                  // stride hi, tensor_dim1_stride lo
        g1[7] = 0;

        int32x4_t gz = {0, 0, 0, 0};
#if defined(__clang_major__) && (__clang_major__ >= 23)
        int32x8_t gz8 = {0, 0, 0, 0, 0, 0, 0, 0};
        __builtin_amdgcn_tensor_load_to_lds(g0, g1, gz, gz, gz8, 0);
#else
        __builtin_amdgcn_tensor_load_to_lds(g0, g1, gz, gz, 0);
#endif
        __builtin_amdgcn_s_wait_tensorcnt(0);
    }
#else
    // Fallback: cooperative vectorized copy.
    for (int idx = threadIdx.x; idx < (IN_H * IN_W / 4); idx += NTHR) {
        ((float4*)smem)[idx] = ((const float4*)src)[idx];
    }
#endif
    __syncthreads();

    // ---- Compute: 64 2x2 output blocks per thread ---------------------
    for (int it = 0; it < (IN_H * IN_W) / (2 * NTHR) * 2; ++it) {
        int b  = (it << 8) + threadIdx.x;      // 0 .. 16383
        int by = b >> 7;
        int bx = b & 127;

        const bool rt = (by > 0), rb = (by < IN_H - 1);
        const bool cl = (bx > 0), cr = (bx < IN_W - 1);

        float tmm = ldz(smem, by - 1, bx - 1, rt && cl);
        float tm0 = ldz(smem, by - 1, bx,     rt);
        float tmp = ldz(smem, by - 1, bx + 1, rt && cr);
        float t0m = ldz(smem, by,     bx - 1, cl);
        float t00 = smem[(by << 7) + bx];
        float t0p = ldz(smem, by,     bx + 1, cr);
        float tpm = ldz(smem, by + 1, bx - 1, rb && cl);
        float tp0 = ldz(smem, by + 1, bx,     rb);
        float tpp = ldz(smem, by + 1, bx + 1, rb && cr);

        // out(y0,x0): taps (i,j) in {0,2}x{0,2} -> weights K[3-i][3-j]
        float o00 = W[15]*tmm + W[13]*tm0 + W[7]*t0m + W[5]*t00;
        float o01 = W[14]*tm0 + W[12]*tmp + W[6]*t00 + W[4]*t0p;
        float o10 = W[11]*t0m + W[9]*t00  + W[3]*tpm + W[1]*tp0;
        float o11 = W[10]*t00 + W[8]*t0p  + W[2]*tp0 + W[0]*tpp;

        int y0 = by << 1;
        int x0 = bx << 1;
        float* p0 = dst + (size_t)y0 * OUT_W + x0;   // y0 even -> even offset in image; 8B aligned only if img even

        if (bx < IN_W - 1) {
            float2 v; v.x = o00; v.y = o01;
            *(float2*)p0 = v;
        } else {
            *p0 = o00;                               // x0 = 254 only
        }
        if (by < IN_H - 1) {
            float* p1 = p0 + OUT_W;                  // odd row: 4B aligned
            if (bx < IN_W - 1) {
                F2u v; v.a = o10; v.b = o11;
                *(F2u*)p1 = v;
            } else {
                *p1 = o10;
            }
        }
    }
}

extern "C" void kernel_launch(void* const* d_in, const int* in_sizes, int n_in,
                              void* d_out, int out_size, void* d_ws, size_t ws_size,
                              hipStream_t stream) {
    const float* x    = (const float*)d_in[0];
    const float* kern = (const float*)d_in[1];
    float*       out  = (float*)d_out;

    const int nc = in_sizes[0] / (IN_H * IN_W);      // 16*64 = 1024 images
    upfirdn2d_up2_k4<<<nc, NTHR, 0, stream>>>(x, kern, out);
}